// LengthRegulator_70824010711579
// MI455X (gfx1250) — compile-verified
//
#include <hip/hip_runtime.h>
#include <hip/hip_bf16.h>
#include <stdint.h>

// Problem constants (from reference)
#define B_       32
#define T_       512
#define D_       384
#define M_       4096
#define D4_      (D_ / 4)          // 96 float4 per row
#define FRAMES_  32                // output frames per block
#define THREADS_ 512               // 16 waves (wave32)

typedef float v4f __attribute__((ext_vector_type(4)));

// ---- async Global->LDS DMA of 2048 B (512 ints) : 128 lanes x b128 ----
__device__ __forceinline__ void async_load_row_2k(const int* gbase, int* lds_dst, int tid) {
    if (tid < (T_ * 4 / 16)) {
        unsigned lds_off = (unsigned)(uintptr_t)(lds_dst + tid * 4); // low 32 bits of flat = LDS offset
        unsigned voff    = (unsigned)tid * 16u;
        asm volatile("global_load_async_to_lds_b128 %0, %1, %2"
                     :: "v"(lds_off), "v"(voff), "s"(gbase)
                     : "memory");
    }
    asm volatile("s_wait_asynccnt 0" ::: "memory");
    __syncthreads();
}

// ---- inclusive scan of clamped durations already staged in sc[0] ------
// Returns pointer to the final buffer (valid for all threads).
__device__ __forceinline__ const int* scan512(int (*sc)[T_], int tid) {
    int v0 = sc[0][tid];
    sc[1][tid] = v0 > 0 ? v0 : 0;
    __syncthreads();
    int src = 1;
#pragma unroll
    for (int off = 1; off < T_; off <<= 1) {
        int v = sc[src][tid];
        if (tid >= off) v += sc[src][tid - off];
        sc[src ^ 1][tid] = v;
        __syncthreads();
        src ^= 1;
    }
    return sc[src];   // 9 steps (odd) -> sc[0]
}

// ---- shared tail: searchsorted + streaming broadcast copy -------------
__device__ __forceinline__ void expand_tail(const float* __restrict__ x,
                                            const int* __restrict__ cum,
                                            float* __restrict__ out,
                                            int b, int t0, int tid,
                                            int* s_idx, int* s_val) {
    const int melLen = cum[T_ - 1];
    const v4f* __restrict__ xb = (const v4f*)(x + (size_t)b * T_ * D_);

    if (tid < FRAMES_) {
        const int t = t0 + tid;
        int lo = 0, hi = T_;
        while (lo < hi) {
            int mid = (lo + hi) >> 1;
            if (cum[mid] <= t) lo = mid + 1; else hi = mid;
        }
        const int idx = lo < T_ ? lo : (T_ - 1);
        const int ok  = (t < melLen) ? 1 : 0;
        s_idx[tid] = idx;
        s_val[tid] = ok;
        if (ok) __builtin_prefetch(&xb[(size_t)idx * D4_], 0, 3); // global_prefetch_b8
    }
    __syncthreads();

    // 32 frames * 96 float4 = 3072 contiguous b128 NT stores per block.
    v4f* __restrict__ ob = (v4f*)(out + ((size_t)b * M_ + t0) * D_);
#pragma unroll
    for (int k = 0; k < (FRAMES_ * D4_) / THREADS_; ++k) {   // 6 iterations
        const int c = tid + k * THREADS_;
        const int f = c / D4_;
        const int q = c - f * D4_;
        v4f v = {0.0f, 0.0f, 0.0f, 0.0f};
        if (s_val[f]) v = xb[(size_t)s_idx[f] * D4_ + q];
        __builtin_nontemporal_store(v, ob + c);              // global_store_b128 th:NT
    }
}

// =============== Kernel 1: per-batch scan -> ws + mel_len ==============
__global__ __launch_bounds__(THREADS_)
void lenreg_scan_kernel(const int* __restrict__ dur,
                        int*       __restrict__ cum_ws,
                        float*     __restrict__ out)
{
    __shared__ int sc[2][T_];
    const int tid = threadIdx.x;
    const int b   = blockIdx.x;

    async_load_row_2k(dur + (size_t)b * T_, &sc[0][0], tid);
    const int* cum = scan512(sc, tid);

    cum_ws[(size_t)b * T_ + tid] = cum[tid];
    if (tid == 0) {
        ((long long*)(out + (size_t)B_ * M_ * D_))[b] = (long long)cum[T_ - 1];
    }
}

// =============== Kernel 2: expand using precomputed cum ================
__global__ __launch_bounds__(THREADS_)
void lenreg_expand_pre_kernel(const float* __restrict__ x,
                              const int*   __restrict__ cum_g,
                              float*       __restrict__ out)
{
    __shared__ int s_cum[T_];
    __shared__ int s_idx[FRAMES_];
    __shared__ int s_val[FRAMES_];

    const int tid = threadIdx.x;
    const int b   = blockIdx.y;
    const int t0  = blockIdx.x * FRAMES_;

    async_load_row_2k(cum_g + (size_t)b * T_, &s_cum[0], tid);
    expand_tail(x, s_cum, out, b, t0, tid, s_idx, s_val);
}

// =============== Fallback: fused scan + expand (no ws needed) ==========
__global__ __launch_bounds__(THREADS_)
void lenreg_expand_fused_kernel(const float* __restrict__ x,
                                const int*   __restrict__ dur,
                                float*       __restrict__ out)
{
    __shared__ int sc[2][T_];
    __shared__ int s_idx[FRAMES_];
    __shared__ int s_val[FRAMES_];

    const int tid = threadIdx.x;
    const int b   = blockIdx.y;
    const int t0  = blockIdx.x * FRAMES_;

    async_load_row_2k(dur + (size_t)b * T_, &sc[0][0], tid);
    const int* cum = scan512(sc, tid);

    if (blockIdx.x == 0 && tid == 0) {
        ((long long*)(out + (size_t)B_ * M_ * D_))[b] = (long long)cum[T_ - 1];
    }
    expand_tail(x, cum, out, b, t0, tid, s_idx, s_val);
}

extern "C" void kernel_launch(void* const* d_in, const int* in_sizes, int n_in,
                              void* d_out, int out_size, void* d_ws, size_t ws_size,
                              hipStream_t stream) {
    (void)in_sizes; (void)n_in; (void)out_size;
    const float* x   = (const float*)d_in[0];
    const int*   dur = (const int*)d_in[1];
    // d_in[2] is max_mel_len (== 4096, compile-time constant here)
    float* out = (float*)d_out;

    const dim3 grid(M_ / FRAMES_, B_);
    const size_t need = (size_t)B_ * T_ * sizeof(int);   // 64 KB for cum

    if (d_ws != nullptr && ws_size >= need) {
        int* cum_ws = (int*)d_ws;
        lenreg_scan_kernel<<<dim3(B_), dim3(THREADS_), 0, stream>>>(dur, cum_ws, out);
        lenreg_expand_pre_kernel<<<grid, dim3(THREADS_), 0, stream>>>(x, cum_ws, out);
    } else {
        lenreg_expand_fused_kernel<<<grid, dim3(THREADS_), 0, stream>>>(x, dur, out);
    }
}